// BicliqueAttentionLayer_35330400977425
// MI455X (gfx1250) — compile-verified
//
#include <hip/hip_runtime.h>
#include <hip/hip_bf16.h>
#include <cstdint>

#define IN_DIM 128
#define OUT_DIM 64
#define NHEAD 4
#define HD 16
#define NEG_SLOPE 0.01f

typedef __attribute__((ext_vector_type(2))) float v2f;
typedef __attribute__((ext_vector_type(8))) float v8f;

__device__ __forceinline__ unsigned flipf(float f) {
  unsigned u = __float_as_uint(f);
  return (u & 0x80000000u) ? ~u : (u | 0x80000000u);
}
__device__ __forceinline__ float unflipf(unsigned u) {
  return __uint_as_float((u & 0x80000000u) ? (u & 0x7FFFFFFFu) : ~u);
}
__device__ __forceinline__ float lrelu(float s) { return s >= 0.f ? s : NEG_SLOPE * s; }

// ---------------- Kernel 1: gumbel-softmax mask over IN_DIM ----------------
__global__ void mask_kernel(const float* __restrict__ logits,
                            const float* __restrict__ gumbel,
                            float* __restrict__ mask) {
  __shared__ float red[IN_DIM];
  int i = threadIdx.x;
  float v = (logits[i] + gumbel[i]) * 2.0f;  // divide by TEMP = 0.5
  red[i] = v;
  __syncthreads();
  for (int off = IN_DIM / 2; off > 0; off >>= 1) {
    if (i < off) red[i] = fmaxf(red[i], red[i + off]);
    __syncthreads();
  }
  float m = red[0];
  __syncthreads();
  float ex = __expf(v - m);
  red[i] = ex;
  __syncthreads();
  for (int off = IN_DIM / 2; off > 0; off >>= 1) {
    if (i < off) red[i] += red[i + off];
    __syncthreads();
  }
  mask[i] = ex / red[0];
}

// ---------------- Kernel 2: h = (feat (.) mask) @ W via V_WMMA_F32_16X16X4_F32 ----
// Block = 128 threads = 4 waves; each wave computes one 16-row x 64-col tile.
// W (with mask folded into rows) staged in LDS in K-pair-major v2f layout so each
// B fragment is a single aligned ds_load_b64 into a contiguous even VGPR pair.
__global__ __launch_bounds__(128) void gemm_wmma_kernel(
    const float* __restrict__ feat, const float* __restrict__ W,
    const float* __restrict__ mask, float* __restrict__ h, int N) {
  __shared__ v2f Wp[(IN_DIM / 2) * OUT_DIM];  // Wp[k/2][col] = {Wm[k][col], Wm[k+1][col]}
  const int tid = threadIdx.x;
  for (int i = tid; i < IN_DIM * OUT_DIM; i += 128) {
    int k = i >> 6;   // row of W
    int n = i & 63;   // col of W
    ((float*)Wp)[(((k >> 1) * OUT_DIM) + n) * 2 + (k & 1)] = W[i] * mask[k];
  }
  __syncthreads();

  const int lane = tid & 31;
  const int wave = tid >> 5;
  const int hf = lane >> 4;   // lane half -> K pair selector
  const int lm = lane & 15;   // M index (A rows) / N index (B,C,D cols)
  const int tileRow = (blockIdx.x * 4 + wave) * 16;
  if (tileRow >= N) return;   // whole wave exits together (EXEC stays all-1 for WMMA)

  v8f acc[NHEAD] = {};
  // A fragment: VGPR r, half hf holds feat[tileRow+lm][k0 + 2*hf + r]
  const float* arow = feat + (size_t)(tileRow + lm) * IN_DIM + 2 * hf;

  for (int k0 = 0; k0 < IN_DIM; k0 += 4) {
    v2f a;
    a.x = arow[k0];
    a.y = arow[k0 + 1];
    // B fragment for head t: K pair (k0/2 + hf), cols t*16 + lm
    const v2f* wp = Wp + ((k0 >> 1) + hf) * OUT_DIM + lm;
#pragma unroll
    for (int t = 0; t < NHEAD; ++t) {
      acc[t] = __builtin_amdgcn_wmma_f32_16x16x4_f32(
          /*neg_a=*/false, a, /*neg_b=*/false, wp[t * HD],
          /*c_mod=*/(short)0, acc[t], /*reuse_a=*/false, /*reuse_b=*/false);
    }
  }

  // C/D layout: VGPR r, half hf, lane lm -> (row tileRow + r + 8*hf, col t*16 + lm)
#pragma unroll
  for (int t = 0; t < NHEAD; ++t) {
#pragma unroll
    for (int r = 0; r < 8; ++r) {
      h[(size_t)(tileRow + r + 8 * hf) * OUT_DIM + t * HD + lm] = acc[t][r];
    }
  }
}

// ---------------- Kernel 3: per-node attention logits (GAT factorization) ------
__global__ void alpha_kernel(const float* __restrict__ h,
                             const float* __restrict__ attn_w,
                             float* __restrict__ asrc, float* __restrict__ adst,
                             int N) {
  int i = blockIdx.x * blockDim.x + threadIdx.x;  // node*NHEAD + head
  if (i >= N * NHEAD) return;
  int head = i & 3;
  int node = i >> 2;
  const float* hv = h + (size_t)node * OUT_DIM + head * HD;
  const float* aw = attn_w + head * 2 * HD;
  float s0 = 0.f, s1 = 0.f;
#pragma unroll
  for (int d = 0; d < HD; ++d) {
    s0 += hv[d] * aw[d];
    s1 += hv[d] * aw[HD + d];
  }
  asrc[i] = s0;
  adst[i] = s1;
}

// ---------------- Kernel 4: init out / denom / smax ----------------
__global__ void init_kernel(float* __restrict__ out, float* __restrict__ denom,
                            unsigned* __restrict__ smax, int N) {
  int i = blockIdx.x * blockDim.x + threadIdx.x;
  if (i < N * OUT_DIM) out[i] = 0.f;
  if (i < N * NHEAD) {
    denom[i] = 0.f;
    smax[i] = 0x007FFFFFu;  // flipf(-inf)
  }
}

// ---------------- Kernel 5: segment max over dst (uint-encoded float max) ------
__global__ void edge_max_kernel(const int* __restrict__ src, const int* __restrict__ dst,
                                const float4* __restrict__ asrc4,
                                const float4* __restrict__ adst4,
                                unsigned* __restrict__ smax, int E) {
  int e = blockIdx.x * blockDim.x + threadIdx.x;
  if (e >= E) return;
  int sN = src[e], dN = dst[e];
  float4 as = asrc4[sN];
  float4 ad = adst4[dN];
  unsigned* sm = smax + (size_t)dN * NHEAD;
  atomicMax(sm + 0, flipf(lrelu(as.x + ad.x)));
  atomicMax(sm + 1, flipf(lrelu(as.y + ad.y)));
  atomicMax(sm + 2, flipf(lrelu(as.z + ad.z)));
  atomicMax(sm + 3, flipf(lrelu(as.w + ad.w)));
}

// ---------------- Kernel 6: segment sum of exp(s - smax) ----------------
__global__ void edge_sum_kernel(const int* __restrict__ src, const int* __restrict__ dst,
                                const float4* __restrict__ asrc4,
                                const float4* __restrict__ adst4,
                                const uint4* __restrict__ smax4,
                                float* __restrict__ denom, int E) {
  int e = blockIdx.x * blockDim.x + threadIdx.x;
  if (e >= E) return;
  int sN = src[e], dN = dst[e];
  float4 as = asrc4[sN];
  float4 ad = adst4[dN];
  uint4 m4 = smax4[dN];
  float* dn = denom + (size_t)dN * NHEAD;
  atomicAdd(dn + 0, __expf(lrelu(as.x + ad.x) - unflipf(m4.x)));
  atomicAdd(dn + 1, __expf(lrelu(as.y + ad.y) - unflipf(m4.y)));
  atomicAdd(dn + 2, __expf(lrelu(as.z + ad.z) - unflipf(m4.z)));
  atomicAdd(dn + 3, __expf(lrelu(as.w + ad.w) - unflipf(m4.w)));
}

// ---------------- Kernel 7: weighted scatter out[dst] += a * h[src] ------------
// 16 threads per edge; each thread handles 4 consecutive channels (float4 load of
// h[src], 4 atomic f32 adds). Attention weight recomputed per thread (cache-hot).
__global__ void scatter_kernel(const int* __restrict__ src, const int* __restrict__ dst,
                               const float* __restrict__ asrc, const float* __restrict__ adst,
                               const unsigned* __restrict__ smax,
                               const float* __restrict__ denom,
                               const float* __restrict__ h, float* __restrict__ out,
                               int E) {
  long long gid = (long long)blockIdx.x * blockDim.x + threadIdx.x;
  int e = (int)(gid >> 4);
  if (e >= E) return;
  int q = (int)(gid & 15);  // quad index: channels [4q, 4q+4)
  int head = q >> 2;
  int sN = src[e], dN = dst[e];
  float s = lrelu(asrc[sN * NHEAD + head] + adst[dN * NHEAD + head]);
  float m = unflipf(smax[dN * NHEAD + head]);
  float a = __expf(s - m) / denom[dN * NHEAD + head];
  float4 hv = *(const float4*)(h + (size_t)sN * OUT_DIM + q * 4);
  float* o = out + (size_t)dN * OUT_DIM + q * 4;
  atomicAdd(o + 0, hv.x * a);
  atomicAdd(o + 1, hv.y * a);
  atomicAdd(o + 2, hv.z * a);
  atomicAdd(o + 3, hv.w * a);
}

extern "C" void kernel_launch(void* const* d_in, const int* in_sizes, int n_in,
                              void* d_out, int out_size, void* d_ws, size_t ws_size,
                              hipStream_t stream) {
  const float* feat   = (const float*)d_in[0];
  const int*   src    = (const int*)d_in[1];
  const int*   dst    = (const int*)d_in[2];
  const float* gumbel = (const float*)d_in[3];
  const float* logits = (const float*)d_in[4];
  const float* W      = (const float*)d_in[5];
  const float* attn_w = (const float*)d_in[6];
  float* out = (float*)d_out;

  const int N = in_sizes[0] / IN_DIM;
  const int E = in_sizes[1];

  // Workspace layout (floats): mask | h | asrc | adst | smax(u32) | denom  (~32 MB)
  // All segments are 16-byte aligned (offsets 128, +N*64, +N*4 ... with N*4, N*64
  // divisible by 4 floats).
  float* ws   = (float*)d_ws;
  float* mask = ws;
  float* h    = mask + IN_DIM;
  float* asrc = h + (size_t)N * OUT_DIM;
  float* adst = asrc + (size_t)N * NHEAD;
  unsigned* smax = (unsigned*)(adst + (size_t)N * NHEAD);
  float* denom = (float*)(smax + (size_t)N * NHEAD);

  mask_kernel<<<1, IN_DIM, 0, stream>>>(logits, gumbel, mask);

  int tiles = (N + 15) / 16;
  gemm_wmma_kernel<<<(tiles + 3) / 4, 128, 0, stream>>>(feat, W, mask, h, N);

  alpha_kernel<<<(N * NHEAD + 255) / 256, 256, 0, stream>>>(h, attn_w, asrc, adst, N);

  init_kernel<<<(N * OUT_DIM + 255) / 256, 256, 0, stream>>>(out, denom, smax, N);

  edge_max_kernel<<<(E + 255) / 256, 256, 0, stream>>>(
      src, dst, (const float4*)asrc, (const float4*)adst, smax, E);

  edge_sum_kernel<<<(E + 255) / 256, 256, 0, stream>>>(
      src, dst, (const float4*)asrc, (const float4*)adst, (const uint4*)smax, denom, E);

  long long scatter_threads = (long long)E * 16;
  scatter_kernel<<<(unsigned)((scatter_threads + 255) / 256), 256, 0, stream>>>(
      src, dst, asrc, adst, smax, denom, h, out, E);
}